// LL_custom_58780922413892
// MI455X (gfx1250) — compile-verified
//
#include <hip/hip_runtime.h>

typedef __attribute__((ext_vector_type(2))) float v2f;
typedef __attribute__((ext_vector_type(8))) float v8f;

#define CIN    64
#define COUT   64
#define BATCH  64
#define IMG    32
#define POSN   (IMG*IMG)        // 1024 spatial positions
#define CSLAB  4                // input channels per K-slab
#define KSLAB  (CSLAB*9)        // 36 K per slab
#define NSLAB  (CIN/CSLAB)      // 16 slabs (K total = 576)
#define APITCH (KSLAB+1)        // 37  (LDS bank-conflict padding)
#define BPITCH (COUT+1)         // 65

__global__ void zero_stats_kernel(float* g) {
    g[threadIdx.x] = 0.0f;      // 128 floats: gsum[64], gsq[64]
}

// One block per spatial position (i,j): GEMM C[64x64] = Patch[64x576] * Wt[576x64]
__global__ __launch_bounds__(256)
void ll_wmma_kernel(const float* __restrict__ x,
                    const float* __restrict__ w,
                    const float* __restrict__ bias,
                    float* __restrict__ y,
                    float* __restrict__ gsum,
                    float* __restrict__ gsq) {
    __shared__ float Alds[BATCH * APITCH];   // patches  [b][kk]
    __shared__ float Blds[KSLAB * BPITCH];   // weights  [kk][o]
    __shared__ float sSum[COUT];
    __shared__ float sSq[COUT];

    const int pos  = blockIdx.x;             // i*32 + j
    const int i    = pos >> 5;
    const int j    = pos & 31;
    const int tid  = threadIdx.x;
    const int lane = tid & 31;               // wave32
    const int wave = tid >> 5;               // 8 waves
    const int half = lane >> 4;
    const int l16  = lane & 15;

    // tile assignment: wave -> (mb, nb..nb+1), 4x4 tile grid of 16x16
    const int mb   = wave & 3;
    const int nb   = (wave >> 2) << 1;
    const int arow = (mb * 16 + l16) * APITCH;
    const int n0   = nb * 16 + l16;          // output channel o for tile 0
    const int n1   = n0 + 16;                // output channel o for tile 1

    if (tid < COUT) { sSum[tid] = 0.0f; sSq[tid] = 0.0f; }

    v8f acc0 = {};
    v8f acc1 = {};

    for (int s = 0; s < NSLAB; ++s) {
        const int c0 = s * CSLAB;

        // ---- stage patch slab A[64][36] into LDS (zero-padded 3x3 taps) ----
        #pragma unroll
        for (int r = 0; r < 9; ++r) {
            int idx = tid + r * 256;                  // 0..2303
            int b   = idx / KSLAB;
            int kk  = idx - b * KSLAB;                // c_local*9 + u*3 + v
            int c   = c0 + kk / 9;
            int uv  = kk % 9;
            int hi  = i + uv / 3 - 1;
            int wj  = j + (uv % 3) - 1;
            float val = 0.0f;
            if ((unsigned)hi < 32u && (unsigned)wj < 32u)
                val = x[(((size_t)b * CIN + c) << 10) + hi * IMG + wj];
            Alds[b * APITCH + kk] = val;
        }

        // ---- stage weight slab Wt[36][64] into LDS (9 contiguous taps/thread) ----
        {
            int o  = tid & 63;
            int cl = tid >> 6;                        // 0..3
            const float* src = w + (size_t)o * (CIN * POSN * 9)
                                 + (size_t)(c0 + cl) * (POSN * 9)
                                 + (size_t)pos * 9;
            #pragma unroll
            for (int uv = 0; uv < 9; ++uv)
                Blds[(cl * 9 + uv) * BPITCH + o] = src[uv];
            if (s + 1 < NSLAB)                        // -> global_prefetch_b8
                __builtin_prefetch(src + (size_t)CSLAB * (POSN * 9), 0, 1);
        }
        __syncthreads();

        // ---- 9 x (K=4) WMMA steps over the slab ----
        #pragma unroll
        for (int k0 = 0; k0 < KSLAB; k0 += 4) {
            int ka = k0 + 2 * half;                   // per-half K offset (ISA A layout)
            v2f a, b0, b1;
            a.x  = Alds[arow + ka];
            a.y  = Alds[arow + ka + 1];
            b0.x = Blds[ka * BPITCH + n0];
            b0.y = Blds[(ka + 1) * BPITCH + n0];
            b1.x = Blds[ka * BPITCH + n1];
            b1.y = Blds[(ka + 1) * BPITCH + n1];
            acc0 = __builtin_amdgcn_wmma_f32_16x16x4_f32(false, a, false, b0,
                                                         (short)0, acc0, false, false);
            acc1 = __builtin_amdgcn_wmma_f32_16x16x4_f32(false, a, false, b1,
                                                         (short)0, acc1, false, false);
        }
        __syncthreads();
    }

    // ---- epilogue: +bias, store pre-BN y, per-channel sum / sumsq ----
    // D layout: lane holds column n, vgpr r -> row (half*8 + r)
    float bv0 = bias[n0 * POSN + pos];
    float bv1 = bias[n1 * POSN + pos];
    float s10 = 0.f, s20 = 0.f, s11 = 0.f, s21 = 0.f;
    #pragma unroll
    for (int r = 0; r < 8; ++r) {
        int bg = mb * 16 + half * 8 + r;             // batch index
        float v0 = acc0[r] + bv0;
        float v1 = acc1[r] + bv1;
        y[(((size_t)bg * COUT + n0) << 10) + pos] = v0;
        y[(((size_t)bg * COUT + n1) << 10) + pos] = v1;
        s10 += v0; s20 += v0 * v0;
        s11 += v1; s21 += v1 * v1;
    }
    atomicAdd(&sSum[n0], s10); atomicAdd(&sSq[n0], s20);   // ds_add_f32
    atomicAdd(&sSum[n1], s11); atomicAdd(&sSq[n1], s21);
    __syncthreads();
    if (tid < COUT) {
        atomicAdd(&gsum[tid], sSum[tid]);                  // global_atomic_add_f32
        atomicAdd(&gsq[tid],  sSq[tid]);
    }
}

// In-place BN (training-mode batch stats) + ReLU over y = d_out
__global__ __launch_bounds__(256)
void bn_relu_kernel(float* __restrict__ y,
                    const float* __restrict__ gsum,
                    const float* __restrict__ gsq,
                    const float* __restrict__ gamma,
                    const float* __restrict__ beta) {
    size_t idx = (size_t)blockIdx.x * 256 + threadIdx.x;   // [b][o][i][j]
    int o = (int)((idx >> 10) & 63);
    const float invN = 1.0f / 65536.0f;                    // N = B*H*W = 64*1024
    float mean = gsum[o] * invN;
    float var  = gsq[o] * invN - mean * mean;
    float sc   = rsqrtf(var + 1e-5f) * gamma[o];
    float v    = (y[idx] - mean) * sc + beta[o];
    y[idx] = fmaxf(v, 0.0f);
}

extern "C" void kernel_launch(void* const* d_in, const int* in_sizes, int n_in,
                              void* d_out, int out_size, void* d_ws, size_t ws_size,
                              hipStream_t stream) {
    const float* x     = (const float*)d_in[0];  // [64,64,32,32]
    const float* w     = (const float*)d_in[1];  // [64,64,32,32,3,3]
    const float* bias  = (const float*)d_in[2];  // [64,32,32]
    const float* gamma = (const float*)d_in[3];  // [64]
    const float* beta  = (const float*)d_in[4];  // [64]
    float* y      = (float*)d_out;               // [64,64,32,32]
    float* gstats = (float*)d_ws;                // 128 floats: sum[64], sumsq[64]

    zero_stats_kernel<<<1, 128, 0, stream>>>(gstats);
    ll_wmma_kernel<<<POSN, 256, 0, stream>>>(x, w, bias, y, gstats, gstats + 64);
    bn_relu_kernel<<<(BATCH * COUT * POSN) / 256, 256, 0, stream>>>(
        y, gstats, gstats + 64, gamma, beta);
}